// Transformer_46789373722799
// MI455X (gfx1250) — compile-verified
//
#include <hip/hip_runtime.h>
#include <math.h>

typedef __attribute__((ext_vector_type(2))) float v2f;
typedef __attribute__((ext_vector_type(8))) float v8f;
typedef __attribute__((ext_vector_type(4))) unsigned int v4u;
typedef __attribute__((ext_vector_type(8))) unsigned int v8u;

#define LSEQ 1024
#define CT   256
#define NB   4
#define NH   16

__device__ __forceinline__ v8f wmma_f32(v2f a, v2f b, v8f c) {
  // V_WMMA_F32_16X16X4_F32 : D(16x16 f32) = A(16x4 f32) * B(4x16 f32) + C
  return __builtin_amdgcn_wmma_f32_16x16x4_f32(false, a, false, b, (short)0, c,
                                               false, false);
}

// Issue a 2D TDM load: tile_rows x 256 f32 tile (row-major, stride 256) from
// `gaddr` into LDS at byte offset `lds_off`. Descriptor layout per
// CDNA5 ISA (D# group0 128b + group1 256b).
__device__ __forceinline__ void tdm_load_2d_256(unsigned lds_off,
                                                unsigned long long gaddr,
                                                unsigned tile_rows,
                                                unsigned tensor_rows) {
  union { unsigned u[4]; v4u v; } G0;
  union { unsigned u[8]; v8u v; } G1;
  G0.u[0] = 1u;                                   // count=1, user descriptor
  G0.u[1] = lds_off;                              // lds_addr (bytes)
  G0.u[2] = (unsigned)(gaddr & 0xffffffffu);      // global_addr[31:0]
  G0.u[3] = (unsigned)((gaddr >> 32) & 0x1ffffffu) | (2u << 30);  // addr[56:32] | type=2
  G1.u[0] = 0x00020000u;          // workgroup_mask=0, data_size=2 (4 bytes)
  G1.u[1] = (unsigned)CT << 16;   // tensor_dim0 = 256 (bits 79:48, low 16)
  G1.u[2] = tensor_rows << 16;    // tensor_dim1 (bits 111:80, low 16)
  G1.u[3] = (unsigned)CT << 16;   // tile_dim0 = 256 (bits 127:112)
  G1.u[4] = tile_rows;            // tile_dim1 (bits 143:128)
  G1.u[5] = (unsigned)CT;         // tensor_dim0_stride = 256 (bits 207:160)
  G1.u[6] = 0u;
  G1.u[7] = 0u;
  asm volatile("tensor_load_to_lds %0, %1" ::"s"(G0.v), "s"(G1.v) : "memory");
}

// ---------------------------------------------------------------------------
// Kernel 1: grouped 1x1 conv  out[n][oc][l] = b[oc] + sum_i x[n][g*32+i][l]*w[g][og][i]
// ---------------------------------------------------------------------------
__global__ void grouped_conv_kernel(const float* __restrict__ x,
                                    const float* __restrict__ w,
                                    const float* __restrict__ bias,
                                    float* __restrict__ out,
                                    int Cout, int Cout_g) {
  int idx = blockIdx.x * blockDim.x + threadIdx.x;
  int total = NB * Cout * LSEQ;
  if (idx >= total) return;
  int l  = idx & (LSEQ - 1);
  int oc = (idx / LSEQ) % Cout;
  int n  = idx / (LSEQ * Cout);
  int g  = oc / Cout_g;
  int og = oc - g * Cout_g;
  const float* xb = x + (size_t)(n * CT + g * 32) * LSEQ + l;
  const float* wb = w + (size_t)(g * Cout_g + og) * 32;
  float acc = bias[oc];
#pragma unroll 8
  for (int i = 0; i < 32; ++i) acc += xb[i * LSEQ] * wb[i];
  out[idx] = acc;
}

// ---------------------------------------------------------------------------
// Kernel 2: flash attention, one wave per (n, h, m-tile of 16 queries).
// S[l,m] = sum_c k[c,l]*q[c,m] ; softmax over l ; O[c,m] = sum_l v[c,l]*P[l,m]
// ---------------------------------------------------------------------------
__global__ void attention_kernel(const float* __restrict__ tokens,
                                 const float* __restrict__ kbuf,
                                 const float* __restrict__ qbuf,
                                 const float* __restrict__ vbuf,
                                 float* __restrict__ x1) {
  const int w    = blockIdx.x;         // 0..4095  (N*H*L/16)
  const int lane = threadIdx.x;        // 0..31
  const int lid  = lane & 15;
  const int half = lane >> 4;
  const int mt = w & 63;
  const int h  = (w >> 6) & 15;
  const int n  = w >> 10;
  const int m0 = mt * 16;

  const float* kb = kbuf + (size_t)(n * 128 + h * 8) * LSEQ;
  const float* qb = qbuf + (size_t)(n * 128 + h * 8) * LSEQ;
  const float* vb = vbuf + (size_t)(n * 256 + h * 16) * LSEQ;

  // B tile of Q (K=c, N=m): lanes<16 hold K=0,1 ; lanes>=16 hold K=2,3
  v2f bq0, bq1;
  {
    int c = 2 * half;
    bq0.x = qb[(c + 0) * LSEQ + m0 + lid];
    bq0.y = qb[(c + 1) * LSEQ + m0 + lid];
    bq1.x = qb[(c + 4) * LSEQ + m0 + lid];
    bq1.y = qb[(c + 5) * LSEQ + m0 + lid];
  }

  v8f o = {0.f, 0.f, 0.f, 0.f, 0.f, 0.f, 0.f, 0.f};
  float m_run = -INFINITY;
  float l_run = 0.f;
  const float scale = 0.03125f;  // 1/sqrt(1024)

  for (int lt = 0; lt < 64; ++lt) {
    const int l0 = lt * 16;
    v2f ak0, ak1;  // A tile of K^T (M = l_local, K = c)
    {
      int c = 2 * half;
      ak0.x = kb[(c + 0) * LSEQ + l0 + lid];
      ak0.y = kb[(c + 1) * LSEQ + l0 + lid];
      ak1.x = kb[(c + 4) * LSEQ + l0 + lid];
      ak1.y = kb[(c + 5) * LSEQ + l0 + lid];
    }
    v8f s = {0.f, 0.f, 0.f, 0.f, 0.f, 0.f, 0.f, 0.f};
    s = wmma_f32(ak0, bq0, s);
    s = wmma_f32(ak1, bq1, s);

    // Online softmax over l for column m = lid (rows split across lane, lane^16)
    float p[8];
    float tmax = -INFINITY;
#pragma unroll
    for (int r = 0; r < 8; ++r) {
      float sv = s[r] * scale;
      p[r] = sv;
      tmax = fmaxf(tmax, sv);
    }
    tmax = fmaxf(tmax, __shfl_xor(tmax, 16, 32));
    float nmax  = fmaxf(m_run, tmax);
    float alpha = __expf(m_run - nmax);
    float psum = 0.f;
#pragma unroll
    for (int r = 0; r < 8; ++r) {
      p[r] = __expf(p[r] - nmax);
      psum += p[r];
    }
    psum += __shfl_xor(psum, 16, 32);
    l_run = l_run * alpha + psum;
    m_run = nmax;
#pragma unroll
    for (int r = 0; r < 8; ++r) o[r] *= alpha;

    // Re-lay P (D layout) into four 4x16 B tiles via cross-half shuffles
    float sp0 = __shfl_xor(p[0], 16, 32), sp1 = __shfl_xor(p[1], 16, 32);
    float sp2 = __shfl_xor(p[2], 16, 32), sp3 = __shfl_xor(p[3], 16, 32);
    float sp4 = __shfl_xor(p[4], 16, 32), sp5 = __shfl_xor(p[5], 16, 32);
    float sp6 = __shfl_xor(p[6], 16, 32), sp7 = __shfl_xor(p[7], 16, 32);
    v2f bp0, bp1, bp2, bp3;
    bp0.x = half ? sp2 : p[0]; bp0.y = half ? sp3 : p[1];  // rows 0..3
    bp1.x = half ? sp6 : p[4]; bp1.y = half ? sp7 : p[5];  // rows 4..7
    bp2.x = half ? p[2] : sp0; bp2.y = half ? p[3] : sp1;  // rows 8..11
    bp3.x = half ? p[6] : sp4; bp3.y = half ? p[7] : sp5;  // rows 12..15

    const float* vrow = vb + (size_t)lid * LSEQ;
#pragma unroll
    for (int kk = 0; kk < 4; ++kk) {
      v2f av;
      int cb = l0 + 4 * kk + 2 * half;
      av.x = vrow[cb];
      av.y = vrow[cb + 1];
      v2f bp = (kk == 0) ? bp0 : (kk == 1) ? bp1 : (kk == 2) ? bp2 : bp3;
      o = wmma_f32(av, bp, o);
    }
  }

  float inv = 1.0f / l_run;
#pragma unroll
  for (int r = 0; r < 8; ++r) {
    int cc = r + 8 * half;
    int m  = m0 + lid;
    size_t idx = (size_t)(n * CT + h * 16 + cc) * LSEQ + m;
    x1[idx] = tokens[idx] + o[r] * inv;
  }
}

// ---------------------------------------------------------------------------
// LayerNorm over channels. Strided input [N,CT,L] -> row-major [P,CT] output.
// ---------------------------------------------------------------------------
__global__ void layernorm_strided_kernel(const float* __restrict__ xin,
                                         const float* __restrict__ g,
                                         const float* __restrict__ be,
                                         float* __restrict__ xout) {
  __shared__ float red[CT];
  int p = blockIdx.x;
  int c = threadIdx.x;
  int n = p >> 10;
  int l = p & 1023;
  float x = xin[(size_t)(n * CT + c) * LSEQ + l];
  red[c] = x;
  __syncthreads();
  for (int s = 128; s > 0; s >>= 1) {
    if (c < s) red[c] += red[c + s];
    __syncthreads();
  }
  float mu = red[0] * (1.0f / CT);
  __syncthreads();
  red[c] = x * x;
  __syncthreads();
  for (int s = 128; s > 0; s >>= 1) {
    if (c < s) red[c] += red[c + s];
    __syncthreads();
  }
  float var = red[0] * (1.0f / CT) - mu * mu;
  xout[(size_t)p * CT + c] = (x - mu) * rsqrtf(var + 1e-5f) * g[c] + be[c];
}

// LayerNorm on row-major [P,CT] input, writing transposed [N,CT,L] output.
__global__ void layernorm_rows_kernel(const float* __restrict__ yin,
                                      const float* __restrict__ g,
                                      const float* __restrict__ be,
                                      float* __restrict__ out) {
  __shared__ float red[CT];
  int p = blockIdx.x;
  int c = threadIdx.x;
  float x = yin[(size_t)p * CT + c];
  red[c] = x;
  __syncthreads();
  for (int s = 128; s > 0; s >>= 1) {
    if (c < s) red[c] += red[c + s];
    __syncthreads();
  }
  float mu = red[0] * (1.0f / CT);
  __syncthreads();
  red[c] = x * x;
  __syncthreads();
  for (int s = 128; s > 0; s >>= 1) {
    if (c < s) red[c] += red[c + s];
    __syncthreads();
  }
  float var = red[0] * (1.0f / CT) - mu * mu;
  int n = p >> 10;
  int l = p & 1023;
  out[(size_t)(n * CT + c) * LSEQ + l] =
      (x - mu) * rsqrtf(var + 1e-5f) * g[c] + be[c];
}

// ---------------------------------------------------------------------------
// FFN GEMMs, 32x32 tile per wave, 8 waves per block covering one 32-row
// stripe. The A stripe (32x256 f32 = 32KB) is DMA'd into LDS once per block
// by the Tensor Data Mover; all 8 waves reuse it. W streams from L2.
// RELU path (GEMM1) and residual path (GEMM2) share the core.
// ---------------------------------------------------------------------------
template <bool RELU>
__global__ void gemm32_kernel(const float* __restrict__ A,   // [P,CT]
                              const float* __restrict__ W,   // [CT,CT]
                              const float* __restrict__ bias,
                              const float* __restrict__ x1,  // residual [N,CT,L] (GEMM2)
                              float* __restrict__ out) {     // [P,CT]
  __shared__ float As[32 * CT];  // 32KB stripe
  const int r0 = blockIdx.x * 32;

  if (threadIdx.x < 32) {  // wave 0 issues the TDM transfer and waits on it
    unsigned lds_off = (unsigned)(unsigned long long)(size_t)As;
    unsigned long long gaddr =
        (unsigned long long)(size_t)(A + (size_t)r0 * CT);
    tdm_load_2d_256(lds_off, gaddr, 32u, (unsigned)(NB * LSEQ));
    __builtin_amdgcn_s_wait_tensorcnt(0);
  }
  __syncthreads();

  const int wave = threadIdx.x >> 5;  // 0..7 -> 32-col block
  const int lane = threadIdx.x & 31;
  const int lid = lane & 15, half = lane >> 4;
  const int o0 = wave * 32;

  const float* a0row = As + (size_t)lid * CT + 2 * half;         // rows 0..15
  const float* a1row = As + (size_t)(16 + lid) * CT + 2 * half;  // rows 16..31
  const float* w0row = W + (size_t)(o0 + lid) * CT + 2 * half;
  const float* w1row = W + (size_t)(o0 + 16 + lid) * CT + 2 * half;

  v8f acc00 = {0.f, 0.f, 0.f, 0.f, 0.f, 0.f, 0.f, 0.f};
  v8f acc01 = acc00, acc10 = acc00, acc11 = acc00;

  for (int k = 0; k < CT; k += 4) {
    v2f a0; a0.x = a0row[k]; a0.y = a0row[k + 1];
    v2f a1; a1.x = a1row[k]; a1.y = a1row[k + 1];
    v2f b0; b0.x = w0row[k]; b0.y = w0row[k + 1];
    v2f b1; b1.x = w1row[k]; b1.y = w1row[k + 1];
    acc00 = wmma_f32(a0, b0, acc00);
    acc01 = wmma_f32(a0, b1, acc01);
    acc10 = wmma_f32(a1, b0, acc10);
    acc11 = wmma_f32(a1, b1, acc11);
  }

  float bv0 = bias[o0 + lid];
  float bv1 = bias[o0 + 16 + lid];
#pragma unroll
  for (int i = 0; i < 2; ++i) {      // row sub-tile
    v8f accA = i ? acc10 : acc00;    // col sub-tile 0
    v8f accB = i ? acc11 : acc01;    // col sub-tile 1
#pragma unroll
    for (int r = 0; r < 8; ++r) {
      int pr = r0 + i * 16 + r + 8 * half;
      int ocA = o0 + lid;
      int ocB = o0 + 16 + lid;
      float vA = accA[r] + bv0;
      float vB = accB[r] + bv1;
      if (RELU) {
        vA = fmaxf(vA, 0.f);
        vB = fmaxf(vB, 0.f);
      } else {
        int n = pr >> 10, l = pr & 1023;
        vA += x1[(size_t)(n * CT + ocA) * LSEQ + l];
        vB += x1[(size_t)(n * CT + ocB) * LSEQ + l];
      }
      out[(size_t)pr * CT + ocA] = vA;
      out[(size_t)pr * CT + ocB] = vB;
    }
  }
}

// ---------------------------------------------------------------------------
extern "C" void kernel_launch(void* const* d_in, const int* in_sizes, int n_in,
                              void* d_out, int out_size, void* d_ws,
                              size_t ws_size, hipStream_t stream) {
  (void)in_sizes; (void)n_in; (void)out_size; (void)ws_size;
  const float* tokens = (const float*)d_in[0];
  const float* wk = (const float*)d_in[1];
  const float* bk = (const float*)d_in[2];
  const float* wq = (const float*)d_in[3];
  const float* bq = (const float*)d_in[4];
  const float* wv = (const float*)d_in[5];
  const float* bv = (const float*)d_in[6];
  const float* w1 = (const float*)d_in[7];
  const float* b1 = (const float*)d_in[8];
  const float* w2 = (const float*)d_in[9];
  const float* b2 = (const float*)d_in[10];
  const float* g1 = (const float*)d_in[11];
  const float* be1 = (const float*)d_in[12];
  const float* g2 = (const float*)d_in[13];
  const float* be2 = (const float*)d_in[14];
  float* out = (float*)d_out;

  float* ws = (float*)d_ws;
  const size_t KQ = (size_t)NB * 128 * LSEQ;   // 524288
  const size_t VV = (size_t)NB * CT * LSEQ;    // 1048576
  float* kbuf = ws;
  float* qbuf = kbuf + KQ;
  float* vbuf = qbuf + KQ;
  float* x1   = vbuf + VV;
  float* xn   = x1 + VV;
  float* hb   = xn + VV;
  float* yb   = hb + VV;

  // projections
  grouped_conv_kernel<<<(NB * 128 * LSEQ + 255) / 256, 256, 0, stream>>>(
      tokens, wk, bk, kbuf, 128, 16);
  grouped_conv_kernel<<<(NB * 128 * LSEQ + 255) / 256, 256, 0, stream>>>(
      tokens, wq, bq, qbuf, 128, 16);
  grouped_conv_kernel<<<(NB * 256 * LSEQ + 255) / 256, 256, 0, stream>>>(
      tokens, wv, bv, vbuf, 256, 32);

  // attention + residual  (N*H*L/16 = 4096 waves)
  attention_kernel<<<4096, 32, 0, stream>>>(tokens, kbuf, qbuf, vbuf, x1);

  // LN1 -> [P,CT]
  layernorm_strided_kernel<<<NB * LSEQ, CT, 0, stream>>>(x1, g1, be1, xn);

  // FFN: 128 blocks x 8 waves, 32x32 tile/wave, TDM-staged A stripe in LDS
  gemm32_kernel<true><<<128, 256, 0, stream>>>(xn, w1, b1, nullptr, hb);
  gemm32_kernel<false><<<128, 256, 0, stream>>>(hb, w2, b2, x1, yb);

  // LN2 -> output [N,CT,L]
  layernorm_rows_kernel<<<NB * LSEQ, CT, 0, stream>>>(yb, g2, be2, out);
}